// EnhancedGNN_20469814133009
// MI455X (gfx1250) — compile-verified
//
#include <hip/hip_runtime.h>
#include <hip/hip_bf16.h>

#define NN   100000
#define EE   1200000
#define BB   64
#define IND  128
#define HID  64
#define OUTD 2
#define EPSV 1e-5f

typedef float v2f __attribute__((ext_vector_type(2)));
typedef float v8f __attribute__((ext_vector_type(8)));

// ---------------------------------------------------------------- utilities
__global__ void zero_kernel(float* __restrict__ p, int n) {
    int i = blockIdx.x * blockDim.x + threadIdx.x;
    if (i < n) p[i] = 0.0f;
}

__global__ void deg_kernel(const int* __restrict__ dst, float* __restrict__ deg) {
    int e = blockIdx.x * blockDim.x + threadIdx.x;
    if (e < EE) atomicAdd(&deg[dst[e]], 1.0f);
}

__global__ void dinv_kernel(float* __restrict__ deg) {
    int i = blockIdx.x * blockDim.x + threadIdx.x;
    if (i < NN) deg[i] = rsqrtf(deg[i] + 1.0f);   // deg buffer becomes dinv
}

// ---------------------------------------------------------------- WMMA GEMM
// C[n x 64] = X[n x K] @ W[K x 64], f32, wave32 V_WMMA_F32_16X16X4_F32.
// Block = 128 threads (4 waves). Each wave computes one 16x16 tile of one
// 16-row strip; 4 waves cover the 64 output columns. K must be multiple of 4.
__global__ void gemm_wmma_kernel(const float* __restrict__ X,
                                 const float* __restrict__ W,
                                 float* __restrict__ C, int n, int K) {
    int strip = blockIdx.x;
    int row0  = strip * 16;
    if (row0 >= n) return;
    int wave  = threadIdx.x >> 5;      // 0..3 -> output column tile
    int lane  = threadIdx.x & 31;
    int rlo   = lane & 15;             // row / col within tile
    int hi    = lane >> 4;             // lane group selects K pair
    int col0  = wave * 16;

    const float* Xr = X + (size_t)(row0 + rlo) * K;       // A: row rlo
    const float* Wc = W + col0 + rlo;                     // B: col rlo

    v8f c = {};
    for (int k0 = 0; k0 < K; k0 += 4) {
        v2f a, b;
        int ka = k0 + 2 * hi;
        a.x = Xr[ka + 0];
        a.y = Xr[ka + 1];
        b.x = Wc[(size_t)(ka + 0) * HID];
        b.y = Wc[(size_t)(ka + 1) * HID];
        c = __builtin_amdgcn_wmma_f32_16x16x4_f32(
                /*neg_a=*/false, a, /*neg_b=*/false, b,
                /*c_mod=*/(short)0, c, /*reuse_a=*/false, /*reuse_b=*/false);
    }

    // C/D layout: vgpr v holds row (v + 8*hi), col = lane&15
    float* Cr = C + (size_t)row0 * HID + col0 + rlo;
#pragma unroll
    for (int v = 0; v < 8; ++v)
        Cr[(size_t)(v + 8 * hi) * HID] = c[v];
}

// ------------------------------------------------- aggregation (gather/scatter)
// agg[i][f] = hw[i][f]*dinv[i]^2 + bias[f]   (self-loop term + bias, also
// serves as initialization so edge kernel can accumulate on top)
__global__ void selfloop_kernel(const float* __restrict__ hw,
                                const float* __restrict__ dinv,
                                const float* __restrict__ bias,
                                float* __restrict__ agg, int n, int lf) {
    size_t i = (size_t)blockIdx.x * blockDim.x + threadIdx.x;
    size_t total = (size_t)n << lf;
    if (i >= total) return;
    int r = (int)(i >> lf);
    int f = (int)(i & ((1u << lf) - 1u));
    float d = dinv[r];
    agg[i] = hw[i] * d * d + bias[f];
}

// one thread per (edge, feature)
__global__ void edge_agg_kernel(const float* __restrict__ hw,
                                const float* __restrict__ dinv,
                                const int* __restrict__ src,
                                const int* __restrict__ dst,
                                float* __restrict__ agg, int lf) {
    size_t i = (size_t)blockIdx.x * blockDim.x + threadIdx.x;
    size_t total = (size_t)EE << lf;
    if (i >= total) return;
    int e = (int)(i >> lf);
    int f = (int)(i & ((1u << lf) - 1u));
    int s = src[e], d = dst[e];
    float w = dinv[s] * dinv[d];
    atomicAdd(&agg[((size_t)d << lf) + f], hw[((size_t)s << lf) + f] * w);
}

// ---------------------------------------------------------------- batchnorm
// In-place ReLU on t (= agg), accumulate per-column sum / sumsq into stats[128]
// Block: 256 threads covers 64 rows (4 rows/thread-slot), LDS tree reduction,
// then 64 atomics per block.
__global__ void relu_stats_kernel(float* __restrict__ t,
                                  float* __restrict__ stats, int n) {
    __shared__ float red[256];
    int f  = threadIdx.x & 63;
    int rl = threadIdx.x >> 6;
    int rbase = blockIdx.x * 64;
    float s = 0.0f, q = 0.0f;
    for (int r = rbase + rl; r < rbase + 64 && r < n; r += 4) {
        size_t idx = (size_t)r * HID + f;
        float v = t[idx];
        v = v > 0.0f ? v : 0.0f;
        t[idx] = v;
        s += v;
        q += v * v;
    }
    red[threadIdx.x] = s;
    __syncthreads();
    if (threadIdx.x < 64) {
        float tot = red[threadIdx.x] + red[threadIdx.x + 64] +
                    red[threadIdx.x + 128] + red[threadIdx.x + 192];
        atomicAdd(&stats[f], tot);
    }
    __syncthreads();
    red[threadIdx.x] = q;
    __syncthreads();
    if (threadIdx.x < 64) {
        float tot = red[threadIdx.x] + red[threadIdx.x + 64] +
                    red[threadIdx.x + 128] + red[threadIdx.x + 192];
        atomicAdd(&stats[64 + f], tot);
    }
}

// stats[f] <- scale, stats[64+f] <- shift  (single block of 64 threads)
__global__ void bn_finalize_kernel(float* __restrict__ stats,
                                   const float* __restrict__ gamma,
                                   const float* __restrict__ beta, int n) {
    int f = threadIdx.x;
    if (f >= 64) return;
    float inv_n = 1.0f / (float)n;
    float mean  = stats[f] * inv_n;
    float var   = stats[64 + f] * inv_n - mean * mean;
    float scale = gamma[f] * rsqrtf(var + EPSV);
    float shift = beta[f] - mean * scale;
    stats[f]      = scale;
    stats[64 + f] = shift;
}

// h[i][f] = t[i][f]*scale[f] + shift[f] (+ h_old[i][f] if residual)
__global__ void bn_apply_kernel(const float* __restrict__ t,
                                const float* __restrict__ stats,
                                float* __restrict__ h, int n, int use_res) {
    size_t i = (size_t)blockIdx.x * blockDim.x + threadIdx.x;
    size_t total = (size_t)n * HID;
    if (i >= total) return;
    int f = (int)(i & 63u);
    float v = t[i] * stats[f] + stats[64 + f];
    if (use_res) v += h[i];
    h[i] = v;
}

// ------------------------------------------------------------ output head
__global__ void out_gemm_kernel(const float* __restrict__ h,
                                const float* __restrict__ W,
                                float* __restrict__ out, int n) {
    int i = blockIdx.x * blockDim.x + threadIdx.x;
    if (i >= n * OUTD) return;
    int r = i >> 1, o = i & 1;
    const float* hr = h + (size_t)r * HID;
    float acc = 0.0f;
#pragma unroll
    for (int k = 0; k < HID; ++k) acc += hr[k] * W[k * OUTD + o];
    out[i] = acc;
}

__global__ void pool_acc_kernel(const float* __restrict__ aggout,
                                const int* __restrict__ batch,
                                float* __restrict__ pools, int n) {
    int i = blockIdx.x * blockDim.x + threadIdx.x;
    if (i >= n) return;
    int g = batch[i];
    atomicAdd(&pools[g * 2 + 0], aggout[(size_t)i * 2 + 0]);
    atomicAdd(&pools[g * 2 + 1], aggout[(size_t)i * 2 + 1]);
    atomicAdd(&pools[BB * 2 + g], 1.0f);
}

__global__ void pool_final_kernel(const float* __restrict__ pools,
                                  float* __restrict__ out) {
    int i = threadIdx.x;   // 0..127
    if (i >= BB * OUTD) return;
    int g = i >> 1;
    float cnt = pools[BB * 2 + g];
    out[i] = pools[i] / fmaxf(cnt, 1.0f);
}

// ---------------------------------------------------------------- launcher
extern "C" void kernel_launch(void* const* d_in, const int* in_sizes, int n_in,
                              void* d_out, int out_size, void* d_ws, size_t ws_size,
                              hipStream_t stream) {
    (void)in_sizes; (void)n_in; (void)out_size; (void)ws_size;
    const float* x        = (const float*)d_in[0];   // [N, 128]
    const float* W_in     = (const float*)d_in[1];   // [128, 64]
    const float* b_in     = (const float*)d_in[2];   // [64]
    const float* W_h      = (const float*)d_in[3];   // [2, 64, 64]
    const float* b_h      = (const float*)d_in[4];   // [2, 64]
    const float* W_out    = (const float*)d_in[5];   // [64, 2]
    const float* b_out    = (const float*)d_in[6];   // [2]
    const float* bn_gamma = (const float*)d_in[7];   // [3, 64]
    const float* bn_beta  = (const float*)d_in[8];   // [3, 64]
    const int*   edge     = (const int*)d_in[9];     // [2, E]
    const int*   batch    = (const int*)d_in[10];    // [N]
    float* out = (float*)d_out;

    const int* src = edge;
    const int* dst = edge + EE;

    float* ws    = (float*)d_ws;
    float* dinv  = ws;                                // N
    float* h     = dinv + NN;                         // N*64
    float* hw    = h + (size_t)NN * HID;              // N*64
    float* agg   = hw + (size_t)NN * HID;             // N*64
    float* stats = agg + (size_t)NN * HID;            // 128
    float* pools = stats + 128;                       // 192

    const int T = 256;
    auto blk = [](size_t n, int t) { return (unsigned)((n + t - 1) / t); };

    // degrees -> dinv (reuse buffer)
    zero_kernel<<<blk(NN, T), T, 0, stream>>>(dinv, NN);
    deg_kernel<<<blk(EE, T), T, 0, stream>>>(dst, dinv);
    dinv_kernel<<<blk(NN, T), T, 0, stream>>>(dinv);

    const size_t nf = (size_t)NN * HID;
    const size_t ef = (size_t)EE * HID;

    // ----- layer 1: GCNConv(128 -> 64) + ReLU + BN
    gemm_wmma_kernel<<<NN / 16, 128, 0, stream>>>(x, W_in, hw, NN, IND);
    selfloop_kernel<<<blk(nf, T), T, 0, stream>>>(hw, dinv, b_in, agg, NN, 6);
    edge_agg_kernel<<<blk(ef, T), T, 0, stream>>>(hw, dinv, src, dst, agg, 6);
    zero_kernel<<<1, 128, 0, stream>>>(stats, 128);
    relu_stats_kernel<<<blk(NN, 64), 256, 0, stream>>>(agg, stats, NN);
    bn_finalize_kernel<<<1, 64, 0, stream>>>(stats, bn_gamma, bn_beta, NN);
    bn_apply_kernel<<<blk(nf, T), T, 0, stream>>>(agg, stats, h, NN, 0);

    // ----- hidden layers: GCNConv(64 -> 64) + ReLU + BN + residual
    for (int L = 0; L < 2; ++L) {
        gemm_wmma_kernel<<<NN / 16, 128, 0, stream>>>(h, W_h + (size_t)L * HID * HID,
                                                      hw, NN, HID);
        selfloop_kernel<<<blk(nf, T), T, 0, stream>>>(hw, dinv, b_h + L * HID, agg, NN, 6);
        edge_agg_kernel<<<blk(ef, T), T, 0, stream>>>(hw, dinv, src, dst, agg, 6);
        zero_kernel<<<1, 128, 0, stream>>>(stats, 128);
        relu_stats_kernel<<<blk(NN, 64), 256, 0, stream>>>(agg, stats, NN);
        bn_finalize_kernel<<<1, 64, 0, stream>>>(stats, bn_gamma + (L + 1) * HID,
                                                 bn_beta + (L + 1) * HID, NN);
        bn_apply_kernel<<<blk(nf, T), T, 0, stream>>>(agg, stats, h, NN, 1);
    }

    // ----- output layer: GCNConv(64 -> 2), no activation
    const size_t no = (size_t)NN * OUTD;
    const size_t eo = (size_t)EE * OUTD;
    out_gemm_kernel<<<blk(no, T), T, 0, stream>>>(h, W_out, hw, NN);   // hw: N x 2
    selfloop_kernel<<<blk(no, T), T, 0, stream>>>(hw, dinv, b_out, agg, NN, 1);
    edge_agg_kernel<<<blk(eo, T), T, 0, stream>>>(hw, dinv, src, dst, agg, 1);

    // ----- global mean pool
    zero_kernel<<<1, 256, 0, stream>>>(pools, BB * 2 + BB);
    pool_acc_kernel<<<blk(NN, T), T, 0, stream>>>(agg, batch, pools, NN);
    pool_final_kernel<<<1, 128, 0, stream>>>(pools, out);
}